// ACCon_63316407878464
// MI455X (gfx1250) — compile-verified
//
#include <hip/hip_runtime.h>
#include <stdint.h>

typedef __attribute__((ext_vector_type(2))) float v2f;
typedef __attribute__((ext_vector_type(8))) float v8f;

#define NTOT   4096          // total anchors = bsz * n_views
#define BSZ    2048
#define DDIM   128
#define NTILES (NTOT / 16)   // 256 column tiles
#define WAVES  4
#define THREADS (WAVES * 32)
#define LSTRIDE 132          // floats; 132 % 64 == 4 dwords -> conflict-free b64 frag reads
                             // 132*4 = 528 B = 33*16 B -> async b128 writes stay aligned
#define TAUV   1e-6f
#define PI_F   3.14159265358979323846f

// cf row r  ->  feature row (r % BSZ) * 2 + (r / BSZ)   (view-major stacking)
__device__ __forceinline__ int cf_feat_row(int r) {
    return ((r & (BSZ - 1)) << 1) + (r >> 11);
}

__global__ __launch_bounds__(THREADS)
void accon_main(const float* __restrict__ feat,
                const int*   __restrict__ labels,
                float*       __restrict__ row_loss)
{
    __shared__ float As[16 * LSTRIDE];              // block's 16-row A panel (full K)
    __shared__ float Bs[WAVES][2][16 * LSTRIDE];    // per-wave double-buffered B tile
    __shared__ float comb[WAVES][16][5];            // cross-wave combine: m, A, B, C, n

    const int tid  = threadIdx.x;
    const int wv   = tid >> 5;
    const int lane = tid & 31;
    const int hsel = lane >> 4;    // 0: lanes 0-15, 1: lanes 16-31
    const int l16  = lane & 15;

    const int row0 = blockIdx.x * 16;               // this block's global cf row base

    // ---- stage A panel: 16 rows x 128 K into LDS (done once) ----
    for (int it = 0; it < (16 * DDIM / 4) / THREADS; ++it) {
        int idx = tid + it * THREADS;               // float4 index; 32 float4 per row
        int r   = idx >> 5;
        int c4  = idx & 31;
        int fr  = cf_feat_row(row0 + r);
        const float4 v = *(const float4*)(feat + fr * DDIM + (c4 << 2));
        float* dst = &As[r * LSTRIDE + (c4 << 2)];
        dst[0] = v.x; dst[1] = v.y; dst[2] = v.z; dst[3] = v.w;
    }

    // row labels for this lane's 8 output rows (C layout: row = p + 8*hsel)
    int labi[8];
#pragma unroll
    for (int p = 0; p < 8; ++p)
        labi[p] = labels[(row0 + p + 8 * hsel) & (BSZ - 1)];

    // per-row running accumulators, distributed over the 16 column-residue lanes
    float mAcc[8], aAcc[8], bAcc[8], cAcc[8], nAcc[8];
#pragma unroll
    for (int p = 0; p < 8; ++p) {
        mAcc[p] = -1e30f; aAcc[p] = 0.f; bAcc[p] = 0.f; cAcc[p] = 0.f; nAcc[p] = 0.f;
    }

    const float* arow = &As[l16 * LSTRIDE + 2 * hsel];

    // Async-stage a 16x128 column tile into Bs[wv][buf].
    // Tiles are 16-aligned and never cross the view boundary, so feature rows
    // within a tile stride by exactly 2 rows = 1024 B in memory.
    auto stage = [&](int tj, int buf) {
        const int      fr0   = cf_feat_row(tj << 4);
        const char*    gaddr = (const char*)(feat + (size_t)fr0 * DDIM) + (lane << 4);
        const uint32_t lbase = (uint32_t)(uintptr_t)(&Bs[wv][buf][0]) + (lane << 4);
#pragma unroll
        for (int r = 0; r < 16; ++r) {
            // one async op per tile row: 32 lanes x 16 B = 512 B = full f32 row
            asm volatile("global_load_async_to_lds_b128 %0, %1, off"
                         :: "v"(lbase + (uint32_t)(r * 528)),
                            "v"(gaddr + (size_t)r * 1024)
                         : "memory");
        }
    };

    __syncthreads();                 // As visible to all waves

    stage(wv, 0);                    // prime buffer 0 with tile t=0

    // 4 waves split the 256 column tiles; no block barriers in the loop
    for (int t = 0; t < NTILES / WAVES; ++t) {
        const int tj = t * WAVES + wv;

        if (t + 1 < NTILES / WAVES) {
            stage((t + 1) * WAVES + wv, (t + 1) & 1);       // prefetch next tile
            // async loads retire in order: <=16 outstanding => current tile landed
            asm volatile("s_wait_asynccnt 0x10" ::: "memory");
        } else {
            asm volatile("s_wait_asynccnt 0x0" ::: "memory");
        }

        const float* brow = &Bs[wv][t & 1][l16 * LSTRIDE + 2 * hsel];

        // ---- 16x16 f32 tile via V_WMMA_F32_16X16X4_F32, K = 128 ----
        v8f acc = {};
#pragma unroll
        for (int k0 = 0; k0 < DDIM; k0 += 4) {
            v2f av = *(const v2f*)(arow + k0);   // lane: row=l16, k=k0+2*hsel+{0,1}
            v2f bv = *(const v2f*)(brow + k0);   // lane: col=l16, k=k0+2*hsel+{0,1}
            acc = __builtin_amdgcn_wmma_f32_16x16x4_f32(
                false, av, false, bv, (short)0, acc, false, false);
        }

        // ---- fused epilogue: 8 elements per lane ----
        const int col  = (tj << 4) + l16;
        const int labj = labels[col & (BSZ - 1)];
#pragma unroll
        for (int p = 0; p < 8; ++p) {
            const int row = row0 + p + 8 * hsel;
            float ct = acc[p];
            ct = fminf(fmaxf(ct, -1.0f), 1.0f);          // clip cos_theta
            mAcc[p] = fmaxf(mAcc[p], ct);                // row max (includes diag)

            float dist = (float)(labi[p] - labj);
            float phi  = (1.0f - dist * 0.01f) * PI_F;
            float sphi, cphi;
            __sincosf(phi, &sphi, &cphi);
            float st = __fsqrt_rn(1.0f - ct * ct + TAUV);
            float g  = ct * cphi - st * fabsf(sphi);      // negative logit

            bool diag = (row == col);
            bool pos  = (labi[p] == labj) && !diag;
            float ep = __expf(ct);
            float eg = __expf(g);
            aAcc[p] += pos ? ep : 0.0f;                   // A = sum_pos exp(cos)
            cAcc[p] += pos ? ct : 0.0f;                   // C = sum_pos cos
            nAcc[p] += pos ? 1.0f : 0.0f;                 // n = #positives
            bAcc[p] += (!pos && !diag) ? eg : 0.0f;       // B = sum_neg exp(g)
        }
    }

    // ---- reduce the 16 column-residue lanes per half-wave (butterfly) ----
#pragma unroll
    for (int p = 0; p < 8; ++p) {
#pragma unroll
        for (int off = 8; off >= 1; off >>= 1) {
            mAcc[p] = fmaxf(mAcc[p], __shfl_xor(mAcc[p], off, 16));
            aAcc[p] += __shfl_xor(aAcc[p], off, 16);
            bAcc[p] += __shfl_xor(bAcc[p], off, 16);
            cAcc[p] += __shfl_xor(cAcc[p], off, 16);
            nAcc[p] += __shfl_xor(nAcc[p], off, 16);
        }
        if (l16 == 0) {
            int rl = p + 8 * hsel;
            comb[wv][rl][0] = mAcc[p];
            comb[wv][rl][1] = aAcc[p];
            comb[wv][rl][2] = bAcc[p];
            comb[wv][rl][3] = cAcc[p];
            comb[wv][rl][4] = nAcc[p];
        }
    }
    __syncthreads();

    // ---- combine the 4 waves' partials and finalize each row ----
    if (tid < 16) {
        float m  = comb[0][tid][0];
        float A  = comb[0][tid][1];
        float Bv = comb[0][tid][2];
        float C  = comb[0][tid][3];
        float n  = comb[0][tid][4];
#pragma unroll
        for (int w = 1; w < WAVES; ++w) {
            m = fmaxf(m, comb[w][tid][0]);
            A  += comb[w][tid][1];
            Bv += comb[w][tid][2];
            C  += comb[w][tid][3];
            n  += comb[w][tid][4];
        }
        // S = sum_j exp(neg_logit)*(j!=i) = e^{-m} * A + B
        float S    = __expf(-m) * A + Bv;
        float mlpp = (C - n * m - n * __logf(S) + TAUV) / (n + TAUV);
        row_loss[row0 + tid] = -mlpp;
    }
}

__global__ __launch_bounds__(256)
void accon_reduce(const float* __restrict__ row_loss, float* __restrict__ out)
{
    __shared__ float sm[256];
    float s = 0.0f;
    for (int i = threadIdx.x; i < NTOT; i += 256) s += row_loss[i];
    sm[threadIdx.x] = s;
    __syncthreads();
    for (int st = 128; st > 0; st >>= 1) {
        if (threadIdx.x < st) sm[threadIdx.x] += sm[threadIdx.x + st];
        __syncthreads();
    }
    if (threadIdx.x == 0) out[0] = sm[0] * (1.0f / (float)NTOT);
}

extern "C" void kernel_launch(void* const* d_in, const int* in_sizes, int n_in,
                              void* d_out, int out_size, void* d_ws, size_t ws_size,
                              hipStream_t stream)
{
    const float* feat   = (const float*)d_in[0];  // [2048, 2, 128] f32 (L2-normalized)
    const int*   labels = (const int*)d_in[1];    // [2048, 1] int32
    float*       out    = (float*)d_out;          // scalar f32
    float*       ws     = (float*)d_ws;           // 4096 per-row losses (16 KB)

    accon_main<<<NTOT / 16, THREADS, 0, stream>>>(feat, labels, ws);
    accon_reduce<<<1, 256, 0, stream>>>(ws, out);
}